// CausalGraphTransformer_30159260352875
// MI455X (gfx1250) — compile-verified
//
#include <hip/hip_runtime.h>
#include <hip/hip_bf16.h>
#include <math.h>

// ---------------- problem constants (match reference) ----------------
#define NN   50000
#define EE   500000
#define DD   128
#define HH   4
#define CC   32
#define EDIM 32
#define FF   512
#define LL   2

typedef __attribute__((ext_vector_type(2))) float v2f;
typedef __attribute__((ext_vector_type(8))) float v8f;

// ---------------- helpers ----------------
__device__ __forceinline__ unsigned f2ord(float f) {
    int i = __float_as_int(f);
    return (unsigned)(i ^ ((i >> 31) | 0x80000000));
}
__device__ __forceinline__ float ord2f(unsigned k) {
    int i = (k & 0x80000000u) ? (int)(k ^ 0x80000000u) : ~(int)k;
    return __int_as_float(i);
}

// block-wide sum for blockDim.x == 128 (4 wave32)
__device__ __forceinline__ float block_sum128(float v, float* sm) {
    #pragma unroll
    for (int off = 16; off > 0; off >>= 1) v += __shfl_xor(v, off, 32);
    int w = threadIdx.x >> 5;
    if ((threadIdx.x & 31) == 0) sm[w] = v;
    __syncthreads();
    float tot = sm[0] + sm[1] + sm[2] + sm[3];
    __syncthreads();
    return tot;
}

// ---------------- fp32 WMMA GEMM with async-to-LDS double buffering --------------
// Y[rows, DOUT] = X[rows, DIN] @ W[DIN, DOUT] (+bias)(+gelu)
// grid.x = rows/16, grid.y = DOUT/128. block = 256 threads = 8 wave32,
// each wave owns one 16x16 output tile, K swept with V_WMMA_F32_16X16X4_F32.
// X tile and W panel are staged in LDS by GLOBAL_LOAD_ASYNC_TO_LDS_B128 (ASYNCcnt),
// double-buffered so chunk c+1 DMA overlaps chunk c WMMAs.
template<int DIN, int DOUT, int ACT>
__global__ __launch_bounds__(256) void gemm_wmma(
    const float* __restrict__ X, const float* __restrict__ W,
    const float* __restrict__ bias, float* __restrict__ Y)
{
    constexpr int KC  = (DIN < 64) ? DIN : 64;   // K-chunk
    constexpr int NCH = DIN / KC;
    constexpr int PX  = KC + 4;                  // Xs row pad (conflict-free b64 A reads)
    constexpr int PW  = 136;                     // Ws row pad (half-waves 16 banks apart)

    __shared__ float Xs[2][16 * PX];
    __shared__ float Ws[2][KC * PW];

    const int row0     = blockIdx.x * 16;
    const int colpanel = blockIdx.y * 128;
    const int tid      = threadIdx.x;

    const int wave = tid >> 5;
    const int lane = tid & 31;
    const int half = lane >> 4;        // lanes 16-31 hold K=2,3 (ISA 7.12.2)
    const int m    = lane & 15;        // A row / B,C,D column within tile
    const int wcol = wave * 16 + m;    // panel-local output column

    // async-stage one K-chunk (X: 16xKC, W: KCx128) into LDS buffer `buf`
    auto issue = [&](int buf, int kc) {
        #pragma unroll
        for (int q = tid; q < 16 * KC / 4; q += 256) {
            int r = (q * 4) / KC, c = (q * 4) % KC;
            unsigned lds = (unsigned)(size_t)&Xs[buf][r * PX + c];
            const float* g = X + (size_t)(row0 + r) * DIN + kc + c;
            asm volatile("global_load_async_to_lds_b128 %0, %1, off"
                         :: "v"(lds), "v"(g) : "memory");
        }
        #pragma unroll
        for (int q = tid; q < KC * 128 / 4; q += 256) {
            int r = (q * 4) / 128, c = (q * 4) % 128;
            unsigned lds = (unsigned)(size_t)&Ws[buf][r * PW + c];
            const float* g = W + (size_t)(kc + r) * DOUT + colpanel + c;
            asm volatile("global_load_async_to_lds_b128 %0, %1, off"
                         :: "v"(lds), "v"(g) : "memory");
        }
    };

    issue(0, 0);
    v8f acc = {};

    for (int ch = 0; ch < NCH; ++ch) {
        asm volatile("s_wait_asynccnt 0x0" ::: "memory");  // this chunk's DMA landed
        __syncthreads();
        if (ch + 1 < NCH) issue((ch + 1) & 1, (ch + 1) * KC);  // overlap next DMA

        const float* xs = &Xs[ch & 1][0];
        const float* ws = &Ws[ch & 1][0];
        #pragma unroll
        for (int ks = 0; ks < KC; ks += 4) {
            const int ka = ks + 2 * half;
            v2f a, b;
            a[0] = xs[m * PX + ka];          // fuses to ds_load_b64
            a[1] = xs[m * PX + ka + 1];
            b[0] = ws[ka * PW + wcol];
            b[1] = ws[(ka + 1) * PW + wcol];
            acc = __builtin_amdgcn_wmma_f32_16x16x4_f32(
                false, a, false, b, (short)0, acc, false, false);
        }
        __syncthreads();                      // buffer reusable two chunks later
    }

    const int coln = colpanel + wcol;
    const float bv = bias ? bias[coln] : 0.0f;
    #pragma unroll
    for (int g = 0; g < 8; ++g) {
        int row = row0 + g + 8 * half;        // D layout: VGPR g -> M=g / g+8
        float o = acc[g] + bv;
        if (ACT == 1) o = 0.5f * o * (1.0f + erff(o * 0.70710678118654752f));
        Y[(size_t)row * DOUT + coln] = o;
    }
}

// ---------------- init per-layer softmax buffers ----------------
__global__ void init_edge_buffers(unsigned* __restrict__ amaxk,
                                  float* __restrict__ denom,
                                  float* __restrict__ agg, int nh, int nd)
{
    int i = blockIdx.x * blockDim.x + threadIdx.x;
    if (i < nd) agg[i] = 0.0f;
    if (i < nh) { denom[i] = 0.0f; amaxk[i] = 0u; }
}

// ---------------- pass A: alpha + segment max ----------------
// block = 256 threads = 2 edges x (H*C = 128). One wave == one (edge, head).
__global__ __launch_bounds__(256) void edge_alpha_kernel(
    const int* __restrict__ src, const int* __restrict__ dst,
    const float* __restrict__ q, const float* __restrict__ k,
    const float* __restrict__ eemb, float* __restrict__ alpha,
    unsigned* __restrict__ amaxk)
{
    const int e = blockIdx.x * 2 + (threadIdx.x >> 7);
    const int t = threadIdx.x & 127;
    const int s = src[e], d = dst[e];
    float kj = k[(size_t)s * DD + t] + eemb[(size_t)e * DD + t];
    float p  = q[(size_t)d * DD + t] * kj;
    #pragma unroll
    for (int off = 16; off > 0; off >>= 1) p += __shfl_xor(p, off, 32);
    if ((threadIdx.x & 31) == 0) {
        int h = t >> 5;
        float av = p * 0.17677669529663687f;     // 1/sqrt(C), C=32
        alpha[(size_t)e * HH + h] = av;
        atomicMax(&amaxk[(size_t)d * HH + h], f2ord(av));
    }
}

// ---------------- pass B: exp + scatter (denom, agg) ----------------
__global__ __launch_bounds__(256) void edge_scatter_kernel(
    const int* __restrict__ src, const int* __restrict__ dst,
    const float* __restrict__ v, const float* __restrict__ eemb,
    const float* __restrict__ alpha, const unsigned* __restrict__ amaxk,
    float* __restrict__ denom, float* __restrict__ agg)
{
    const int e = blockIdx.x * 2 + (threadIdx.x >> 7);
    const int t = threadIdx.x & 127;
    const int h = t >> 5;
    const int s = src[e], d = dst[e];
    float am = ord2f(amaxk[(size_t)d * HH + h]);
    float ex = expf(alpha[(size_t)e * HH + h] - am);
    if ((threadIdx.x & 31) == 0) atomicAdd(&denom[(size_t)d * HH + h], ex);
    float vj = v[(size_t)s * DD + t] + eemb[(size_t)e * DD + t];
    atomicAdd(&agg[(size_t)d * DD + t], vj * ex);
}

// ---------------- combine: out/denom, beta gate, residual, LayerNorm1 ----------------
__global__ __launch_bounds__(128) void combine_beta_ln_kernel(
    const float* __restrict__ agg, const float* __restrict__ denom,
    const float* __restrict__ r, const float* __restrict__ xin,
    const float* __restrict__ wbeta, const float* __restrict__ g1,
    const float* __restrict__ b1, float* __restrict__ xout)
{
    __shared__ float sm[4];
    const int n = blockIdx.x, t = threadIdx.x, h = t >> 5;
    float out = agg[(size_t)n * DD + t] / (denom[(size_t)n * HH + h] + 1e-16f);
    float rr  = r[(size_t)n * DD + t];
    float contrib = out * wbeta[t] + rr * wbeta[DD + t] + (out - rr) * wbeta[2 * DD + t];
    float sdot = block_sum128(contrib, sm);
    float beta = 1.0f / (1.0f + expf(-sdot));
    float hv = beta * rr + (1.0f - beta) * out;
    float y  = xin[(size_t)n * DD + t] + hv;
    float mean = block_sum128(y, sm) * (1.0f / DD);
    float dy = y - mean;
    float var = block_sum128(dy * dy, sm) * (1.0f / DD);
    xout[(size_t)n * DD + t] = dy * rsqrtf(var + 1e-5f) * g1[t] + b1[t];
}

// ---------------- residual + LayerNorm2 ----------------
__global__ __launch_bounds__(128) void add_ln_kernel(
    const float* __restrict__ a, const float* __restrict__ b,
    const float* __restrict__ g2, const float* __restrict__ b2,
    float* __restrict__ xout)
{
    __shared__ float sm[4];
    const int n = blockIdx.x, t = threadIdx.x;
    float y = a[(size_t)n * DD + t] + b[(size_t)n * DD + t];
    float mean = block_sum128(y, sm) * (1.0f / DD);
    float dy = y - mean;
    float var = block_sum128(dy * dy, sm) * (1.0f / DD);
    xout[(size_t)n * DD + t] = dy * rsqrtf(var + 1e-5f) * g2[t] + b2[t];
}

// ---------------- host side ----------------
extern "C" void kernel_launch(void* const* d_in, const int* in_sizes, int n_in,
                              void* d_out, int out_size, void* d_ws, size_t ws_size,
                              hipStream_t stream)
{
    (void)in_sizes; (void)n_in; (void)out_size; (void)ws_size;

    const float* x0    = (const float*)d_in[0];
    const int*   ei    = (const int*)d_in[1];
    const int*   srcp  = ei;
    const int*   dstp  = ei + EE;
    const float* eattr = (const float*)d_in[2];
    const float* Wq    = (const float*)d_in[3];
    const float* bq    = (const float*)d_in[4];
    const float* Wk    = (const float*)d_in[5];
    const float* bk    = (const float*)d_in[6];
    const float* Wv    = (const float*)d_in[7];
    const float* bv    = (const float*)d_in[8];
    const float* We    = (const float*)d_in[9];
    const float* Wskip = (const float*)d_in[10];
    const float* bskip = (const float*)d_in[11];
    const float* Wbeta = (const float*)d_in[12];
    const float* g1    = (const float*)d_in[13];
    const float* b1    = (const float*)d_in[14];
    const float* g2    = (const float*)d_in[15];
    const float* b2    = (const float*)d_in[16];
    const float* Wff1  = (const float*)d_in[17];
    const float* bff1  = (const float*)d_in[18];
    const float* Wff2  = (const float*)d_in[19];
    const float* bff2  = (const float*)d_in[20];

    float* ws = (float*)d_ws;
    size_t o = 0;
    float* qb    = ws + o; o += (size_t)NN * DD;
    float* kb_   = ws + o; o += (size_t)NN * DD;
    float* vb    = ws + o; o += (size_t)NN * DD;
    float* rb    = ws + o; o += (size_t)NN * DD;
    float* h1b   = qb;                      // alias: N*512 == q..r region (dead by FF time)
    float* eembb = ws + o; o += (size_t)EE * DD;
    float* alphab= ws + o; o += (size_t)EE * HH;
    unsigned* amaxb = (unsigned*)(ws + o); o += (size_t)NN * HH;
    float* denomb= ws + o; o += (size_t)NN * HH;
    float* aggb  = ws + o; o += (size_t)NN * DD;
    float* x1b   = ws + o; o += (size_t)NN * DD;
    float* h2b   = ws + o; o += (size_t)NN * DD;
    float* xbuf  = ws + o; o += (size_t)NN * DD;

    const dim3 blk256(256), blk128(128);

    for (int i = 0; i < LL; ++i) {
        const float* xin  = (i == 0) ? x0 : xbuf;
        float*       xout = (i == LL - 1) ? (float*)d_out : xbuf;

        const size_t wDD = (size_t)i * DD * DD;

        // q, k, v, skip projections  (N x 128 @ 128 x 128)
        gemm_wmma<DD, DD, 0><<<dim3(NN / 16, 1), blk256, 0, stream>>>(xin, Wq + wDD, bq + i * DD, qb);
        gemm_wmma<DD, DD, 0><<<dim3(NN / 16, 1), blk256, 0, stream>>>(xin, Wk + wDD, bk + i * DD, kb_);
        gemm_wmma<DD, DD, 0><<<dim3(NN / 16, 1), blk256, 0, stream>>>(xin, Wv + wDD, bv + i * DD, vb);
        gemm_wmma<DD, DD, 0><<<dim3(NN / 16, 1), blk256, 0, stream>>>(xin, Wskip + wDD, bskip + i * DD, rb);

        // edge embedding  (E x 32 @ 32 x 128, no bias)
        gemm_wmma<EDIM, DD, 0><<<dim3(EE / 16, 1), blk256, 0, stream>>>(eattr, We + (size_t)i * EDIM * DD, nullptr, eembb);

        // segment softmax + aggregation
        init_edge_buffers<<<(NN * DD + 255) / 256, blk256, 0, stream>>>(amaxb, denomb, aggb, NN * HH, NN * DD);
        edge_alpha_kernel<<<EE / 2, blk256, 0, stream>>>(srcp, dstp, qb, kb_, eembb, alphab, amaxb);
        edge_scatter_kernel<<<EE / 2, blk256, 0, stream>>>(srcp, dstp, vb, eembb, alphab, amaxb, denomb, aggb);

        // beta gate + residual + LN1
        combine_beta_ln_kernel<<<NN, blk128, 0, stream>>>(aggb, denomb, rb, xin,
                                                          Wbeta + (size_t)i * 3 * DD,
                                                          g1 + i * DD, b1 + i * DD, x1b);

        // FFN: gelu(x1 @ Wff1 + b) @ Wff2 + b, residual + LN2
        gemm_wmma<DD, FF, 1><<<dim3(NN / 16, 4), blk256, 0, stream>>>(x1b, Wff1 + (size_t)i * DD * FF, bff1 + i * FF, h1b);
        gemm_wmma<FF, DD, 0><<<dim3(NN / 16, 1), blk256, 0, stream>>>(h1b, Wff2 + (size_t)i * FF * DD, bff2 + i * DD, h2b);
        add_ln_kernel<<<NN, blk128, 0, stream>>>(x1b, h2b, g2 + i * DD, b2 + i * DD, xout);
    }
}